// TAMA_s3_20607253086497
// MI455X (gfx1250) — compile-verified
//
#include <hip/hip_runtime.h>
#include <math.h>

#define IN_DIM 320
#define NPTS   2048
#define NSUP   64
#define NCLS   24
#define DATT   12
#define INV_TEMP 0.2886751345948129f   // 1/sqrt(12)
#define LN_EPS   1e-5f
#define SW 33                           // padded LDS row stride (32 cols + 1)

typedef __attribute__((ext_vector_type(2))) float v2f;
typedef __attribute__((ext_vector_type(8))) float v8f;

// workspace layout in floats
#define OFF_Q    0                      // 12*320
#define OFF_K    4096                   // 64*12*320
#define OFF_ATTN 262144                 // 64*320*320 (later aliased as C_n)
#define OFF_T    6815744                // 64*320*320

// ---------------- Kernel 1: q = Wq @ text_emb^T  (12 x 320) ----------------
__global__ void q_kernel(const float* __restrict__ Wq,
                         const float* __restrict__ text,
                         float* __restrict__ q) {
    int i = blockIdx.x * blockDim.x + threadIdx.x;
    if (i >= DATT * IN_DIM) return;
    int d = i / IN_DIM, x = i % IN_DIM;
    float acc = 0.f;
#pragma unroll
    for (int c = 0; c < NCLS; ++c)
        acc += Wq[d * NCLS + c] * text[x * NCLS + c];
    q[d * IN_DIM + x] = acc;
}

// ---------------- Kernel 2: k[n,d,x] = sum_p Wk[d,p] * S[n,x,p] ----------------
// Wk (12x2048 = 96KB) staged in dynamic LDS; one wave per x row; shfl reduce.
__global__ void __launch_bounds__(256) k_kernel(const float* __restrict__ Wk,
                                                const float* __restrict__ S,
                                                float* __restrict__ kout) {
    extern __shared__ float lwk[];                  // 12*2048 floats
    int n = blockIdx.y;
    int xg = blockIdx.x;                            // 0..3 (80 rows each)
    int tid = threadIdx.x;
    for (int i = tid; i < DATT * NPTS; i += 256) lwk[i] = Wk[i];
    __syncthreads();
    int w = tid >> 5, lane = tid & 31;
    const float* Sn = S + (size_t)n * IN_DIM * NPTS;
    for (int x = xg * 80 + w; x < xg * 80 + 80; x += 8) {
        float acc[DATT];
#pragma unroll
        for (int d = 0; d < DATT; ++d) acc[d] = 0.f;
        const float* row = Sn + (size_t)x * NPTS;
        for (int p = lane; p < NPTS; p += 32) {
            float s = row[p];
#pragma unroll
            for (int d = 0; d < DATT; ++d) acc[d] += s * lwk[d * NPTS + p];
        }
#pragma unroll
        for (int d = 0; d < DATT; ++d) {
            float v = acc[d];
            for (int off = 16; off > 0; off >>= 1) v += __shfl_xor(v, off, 32);
            if (lane == 0) kout[((size_t)n * DATT + d) * IN_DIM + x] = v;
        }
    }
}

// ---------------- Kernel 3: attn[n,x,:] = softmax(q[:,x] . k[n,:,:] / sqrt(12)) ----------------
__global__ void attn_kernel(const float* __restrict__ q,
                            const float* __restrict__ k,
                            float* __restrict__ attn) {
    __shared__ float wred[10];
    __shared__ float bval;
    int n = blockIdx.y, x = blockIdx.x;
    int tid = threadIdx.x;                           // y in [0,320)
    int w = tid >> 5, lane = tid & 31;
    const float* kn = k + (size_t)n * DATT * IN_DIM;
    float l = 0.f;
#pragma unroll
    for (int d = 0; d < DATT; ++d)
        l += q[d * IN_DIM + x] * kn[d * IN_DIM + tid];
    l *= INV_TEMP;
    // block max
    float m = l;
    for (int off = 16; off > 0; off >>= 1) m = fmaxf(m, __shfl_xor(m, off, 32));
    if (lane == 0) wred[w] = m;
    __syncthreads();
    if (tid == 0) {
        float mm = wred[0];
        for (int i = 1; i < 10; ++i) mm = fmaxf(mm, wred[i]);
        bval = mm;
    }
    __syncthreads();
    float e = expf(l - bval);
    float s = e;
    for (int off = 16; off > 0; off >>= 1) s += __shfl_xor(s, off, 32);
    __syncthreads();
    if (lane == 0) wred[w] = s;
    __syncthreads();
    if (tid == 0) {
        float ss = 0.f;
        for (int i = 0; i < 10; ++i) ss += wred[i];
        bval = 1.0f / ss;
    }
    __syncthreads();
    attn[((size_t)n * IN_DIM + x) * IN_DIM + tid] = e * bval;
}

// ---------------- Kernel 4: batched 320x320x320 GEMM via V_WMMA_F32_16X16X4_F32 ----------------
// 10 waves/block; each wave register-blocks a 2x2 grid of 16x16 tiles
// (rt = w, w+10 ; ct = 2*bx, 2*bx+1) so A/B fragments are each reused twice.
__global__ void __launch_bounds__(320) gemm320_wmma(const float* __restrict__ Abase, long long sA,
                                                    const float* __restrict__ Bbase, long long sB,
                                                    float* __restrict__ Cbase, long long sC) {
    int n = blockIdx.y;
    int w = threadIdx.x >> 5, lane = threadIdx.x & 31;
    int half = lane >> 4, lm = lane & 15;
    int rt0 = w, rt1 = w + 10;
    int ct0 = blockIdx.x * 2, ct1 = ct0 + 1;
    const float* A = Abase + (size_t)n * sA;
    const float* B = Bbase + (size_t)n * sB;
    float* C = Cbase + (size_t)n * sC;
    const float* arow0 = A + (rt0 * 16 + lm) * IN_DIM;
    const float* arow1 = A + (rt1 * 16 + lm) * IN_DIM;
#pragma unroll
    for (int pk = 0; pk < IN_DIM; pk += 64) {        // global_prefetch_b8
        __builtin_prefetch(arow0 + pk, 0, 0);
        __builtin_prefetch(arow1 + pk, 0, 0);
    }
    v8f acc00 = {}, acc01 = {}, acc10 = {}, acc11 = {};
    for (int k = 0; k < IN_DIM; k += 4) {
        int kk = k + 2 * half;
        v2f a0, a1, b0, b1;
        a0.x = arow0[kk];              a0.y = arow0[kk + 1];
        a1.x = arow1[kk];              a1.y = arow1[kk + 1];
        b0.x = B[kk * IN_DIM + ct0 * 16 + lm];
        b0.y = B[(kk + 1) * IN_DIM + ct0 * 16 + lm];
        b1.x = B[kk * IN_DIM + ct1 * 16 + lm];
        b1.y = B[(kk + 1) * IN_DIM + ct1 * 16 + lm];
        acc00 = __builtin_amdgcn_wmma_f32_16x16x4_f32(false, a0, false, b0, (short)0, acc00, false, false);
        acc01 = __builtin_amdgcn_wmma_f32_16x16x4_f32(false, a0, false, b1, (short)0, acc01, false, false);
        acc10 = __builtin_amdgcn_wmma_f32_16x16x4_f32(false, a1, false, b0, (short)0, acc10, false, false);
        acc11 = __builtin_amdgcn_wmma_f32_16x16x4_f32(false, a1, false, b1, (short)0, acc11, false, false);
    }
#pragma unroll
    for (int r = 0; r < 8; ++r) {
        int row0 = rt0 * 16 + half * 8 + r;
        int row1 = rt1 * 16 + half * 8 + r;
        C[row0 * IN_DIM + ct0 * 16 + lm] = acc00[r];
        C[row0 * IN_DIM + ct1 * 16 + lm] = acc01[r];
        C[row1 * IN_DIM + ct0 * 16 + lm] = acc10[r];
        C[row1 * IN_DIM + ct1 * 16 + lm] = acc11[r];
    }
}

// ---------------- Kernel 5: out = LayerNorm(C_n @ S_n + S_n) fused ----------------
// Block: one n x 32 columns, 10 waves. S tile + result tile in LDS; each wave
// computes a 2x2 grid of 16x16 WMMA tiles (rt = w, w+10; ct = 0,1) over K=320;
// residual + channel-dim LayerNorm done in-LDS; coalesced 128B stores.
__global__ void __launch_bounds__(320) out_ln_wmma(const float* __restrict__ Cm,
                                                   const float* __restrict__ S,
                                                   const float* __restrict__ gamma,
                                                   const float* __restrict__ beta,
                                                   float* __restrict__ out) {
    extern __shared__ float smem[];
    float* sS = smem;                  // IN_DIM*SW
    float* sO = smem + IN_DIM * SW;    // IN_DIM*SW
    float* pS = sO + IN_DIM * SW;      // 320
    float* pQ = pS + 320;              // 320
    float* cM = pQ + 320;              // 32
    float* cI = cM + 32;               // 32

    int n = blockIdx.y;
    int col0 = blockIdx.x * 32;
    int tid = threadIdx.x;
    const float* Sn = S + (size_t)n * IN_DIM * NPTS;
    const float* Cn = Cm + (size_t)n * IN_DIM * IN_DIM;

    // stage S tile (320 rows x 32 cols), padded stride
    for (int i = tid; i < IN_DIM * 32; i += 320) {
        int row = i >> 5, c = i & 31;
        sS[row * SW + c] = Sn[(size_t)row * NPTS + col0 + c];
    }
    __syncthreads();

    int w = tid >> 5, lane = tid & 31, half = lane >> 4, lm = lane & 15;
    int rt0 = w, rt1 = w + 10;
    const float* arow0 = Cn + (rt0 * 16 + lm) * IN_DIM;
    const float* arow1 = Cn + (rt1 * 16 + lm) * IN_DIM;
#pragma unroll
    for (int pk = 0; pk < IN_DIM; pk += 64) {        // global_prefetch_b8
        __builtin_prefetch(arow0 + pk, 0, 0);
        __builtin_prefetch(arow1 + pk, 0, 0);
    }
    v8f acc00 = {}, acc01 = {}, acc10 = {}, acc11 = {};
    for (int k = 0; k < IN_DIM; k += 4) {
        int kk = k + 2 * half;
        v2f a0, a1, b0, b1;
        a0.x = arow0[kk];              a0.y = arow0[kk + 1];
        a1.x = arow1[kk];              a1.y = arow1[kk + 1];
        b0.x = sS[kk * SW + lm];
        b0.y = sS[(kk + 1) * SW + lm];
        b1.x = sS[kk * SW + 16 + lm];
        b1.y = sS[(kk + 1) * SW + 16 + lm];
        acc00 = __builtin_amdgcn_wmma_f32_16x16x4_f32(false, a0, false, b0, (short)0, acc00, false, false);
        acc01 = __builtin_amdgcn_wmma_f32_16x16x4_f32(false, a0, false, b1, (short)0, acc01, false, false);
        acc10 = __builtin_amdgcn_wmma_f32_16x16x4_f32(false, a1, false, b0, (short)0, acc10, false, false);
        acc11 = __builtin_amdgcn_wmma_f32_16x16x4_f32(false, a1, false, b1, (short)0, acc11, false, false);
    }
#pragma unroll
    for (int r = 0; r < 8; ++r) {
        int row0 = rt0 * 16 + half * 8 + r;
        int row1 = rt1 * 16 + half * 8 + r;
        int c0 = lm, c1 = 16 + lm;
        sO[row0 * SW + c0] = acc00[r] + sS[row0 * SW + c0];   // residual
        sO[row0 * SW + c1] = acc01[r] + sS[row0 * SW + c1];
        sO[row1 * SW + c0] = acc10[r] + sS[row1 * SW + c0];
        sO[row1 * SW + c1] = acc11[r] + sS[row1 * SW + c1];
    }
    __syncthreads();

    // column statistics over 320 channels (rows)
    {
        int c = tid & 31, seg = tid >> 5;            // 10 segments x 32 rows
        float s = 0.f, qq = 0.f;
        for (int r = seg * 32; r < seg * 32 + 32; ++r) {
            float v = sO[r * SW + c];
            s += v; qq += v * v;
        }
        pS[tid] = s; pQ[tid] = qq;
    }
    __syncthreads();
    if (tid < 32) {
        float s = 0.f, qq = 0.f;
        for (int g = 0; g < 10; ++g) { s += pS[g * 32 + tid]; qq += pQ[g * 32 + tid]; }
        float mean = s * (1.0f / IN_DIM);
        float var = qq * (1.0f / IN_DIM) - mean * mean;
        cM[tid] = mean;
        cI[tid] = rsqrtf(var + LN_EPS);
    }
    __syncthreads();

    float* On = out + (size_t)n * IN_DIM * NPTS;
    for (int i = tid; i < IN_DIM * 32; i += 320) {
        int row = i >> 5, c = i & 31;
        float v = (sO[row * SW + c] - cM[c]) * cI[c] * gamma[row] + beta[row];
        On[(size_t)row * NPTS + col0 + c] = v;
    }
}

extern "C" void kernel_launch(void* const* d_in, const int* in_sizes, int n_in,
                              void* d_out, int out_size, void* d_ws, size_t ws_size,
                              hipStream_t stream) {
    const float* text  = (const float*)d_in[0];
    const float* S     = (const float*)d_in[1];
    const float* Wq    = (const float*)d_in[2];
    const float* Wk    = (const float*)d_in[3];
    const float* Wv    = (const float*)d_in[4];
    const float* Wfc   = (const float*)d_in[5];
    const float* gamma = (const float*)d_in[6];
    const float* beta  = (const float*)d_in[7];
    float* out = (float*)d_out;
    float* ws  = (float*)d_ws;

    float* q    = ws + OFF_Q;
    float* kbuf = ws + OFF_K;
    float* attn = ws + OFF_ATTN;
    float* Tbuf = ws + OFF_T;
    float* Cmat = attn;                               // attn dead after gemm 4a

    const long long s2 = (long long)IN_DIM * IN_DIM;

    q_kernel<<<dim3((DATT * IN_DIM + 255) / 256), 256, 0, stream>>>(Wq, text, q);
    k_kernel<<<dim3(4, NSUP), 256, DATT * NPTS * sizeof(float), stream>>>(Wk, S, kbuf);
    attn_kernel<<<dim3(IN_DIM, NSUP), IN_DIM, 0, stream>>>(q, kbuf, attn);
    // T_n = attn_n @ Wv
    gemm320_wmma<<<dim3(10, NSUP), 320, 0, stream>>>(attn, s2, Wv, 0, Tbuf, s2);
    // C_n = Wfc @ T_n
    gemm320_wmma<<<dim3(10, NSUP), 320, 0, stream>>>(Wfc, 0, Tbuf, s2, Cmat, s2);
    // out = LN(C_n @ S_n + S_n)
    size_t shm5 = (size_t)(2 * IN_DIM * SW + 2 * 320 + 64) * sizeof(float);
    out_ln_wmma<<<dim3(NPTS / 32, NSUP), 320, shm5, stream>>>(Cmat, S, gamma, beta, out);
}